// PieceVectorExtractor_18184891531343
// MI455X (gfx1250) — compile-verified
//
#include <hip/hip_runtime.h>

typedef __attribute__((ext_vector_type(2))) float v2f;
typedef __attribute__((ext_vector_type(8))) float v8f;

#define WAVES_PER_BLOCK 8

// board: [B, 11, 64] f32, ids: [B, 64] i32, W: [64, 11] f32, bias: [64] f32
// out: [B, 32, 64] f32
__global__ __launch_bounds__(256) void piece_extract_gemm_kernel(
    const float* __restrict__ board,
    const int*   __restrict__ ids,
    const float* __restrict__ W,
    const float* __restrict__ bias,
    float*       __restrict__ out,
    int Btot)
{
    __shared__ alignas(16) int   s_ids[WAVES_PER_BLOCK][64];
    __shared__ alignas(16) float s_feats[WAVES_PER_BLOCK][32][12]; // K padded 11->12

    const int lane = threadIdx.x & 31;
    const int wave = threadIdx.x >> 5;
    int b = blockIdx.x * WAVES_PER_BLOCK + wave;
    if (b >= Btot) b = Btot - 1;   // tail clamp: redundant identical writes, race-free

    // ---- Phase 1: stage this board's 64 piece ids into LDS -----------------
    const int* idp = ids + (size_t)b * 64;
    s_ids[wave][lane]      = idp[lane];
    s_ids[wave][lane + 32] = idp[lane + 32];
    __syncthreads();

    // ---- Phase 2: lane p finds first square holding piece id (p+1) --------
    const int target = lane + 1;
    int first = -1;
    {
        const int4* idv = (const int4*)(&s_ids[wave][0]);
        #pragma unroll
        for (int j = 15; j >= 0; --j) {     // descend so final hit = smallest index
            int4 v = idv[j];
            if (v.w == target) first = j * 4 + 3;
            if (v.z == target) first = j * 4 + 2;
            if (v.y == target) first = j * 4 + 1;
            if (v.x == target) first = j * 4 + 0;
        }
    }
    const bool exists = (first >= 0);
    const int  sq     = exists ? first : 0;

    // ---- Phase 3: gather 11 channel features for this piece into LDS ------
    const float* bp = board + (size_t)b * (11 * 64) + sq;
    float f[12];
    #pragma unroll
    for (int c = 0; c < 11; ++c) {
        float v = bp[c * 64];               // always in-range (sq clamped)
        f[c] = exists ? v : 0.0f;
    }
    f[11] = 0.0f;                           // K padding
    float4* frow = (float4*)(&s_feats[wave][lane][0]);  // 48B rows: 16B aligned
    frow[0] = make_float4(f[0], f[1], f[2],  f[3]);
    frow[1] = make_float4(f[4], f[5], f[6],  f[7]);
    frow[2] = make_float4(f[8], f[9], f[10], f[11]);
    __syncthreads();

    // ---- Phase 4: fragments ------------------------------------------------
    const int l16   = lane & 15;
    const int khalf = (lane >> 4) ? 2 : 0;  // lanes 16-31 hold K+2,K+3 of each tile

    // B fragments: B[kk][o] = W[o][kk] (W.T), o = n*16 + l16; W stays L2-hot
    v2f bfrag[4][3];
    #pragma unroll
    for (int n = 0; n < 4; ++n) {
        const int o = n * 16 + l16;
        #pragma unroll
        for (int k = 0; k < 3; ++k) {
            const int kk = k * 4 + khalf;
            bfrag[n][k].x = (kk     < 11) ? W[o * 11 + kk]     : 0.0f;
            bfrag[n][k].y = (kk + 1 < 11) ? W[o * 11 + kk + 1] : 0.0f;
        }
    }

    // A fragments from LDS: rows = pieces, cols = channels
    v2f afrag[2][3];
    #pragma unroll
    for (int m = 0; m < 2; ++m) {
        const float* fr = &s_feats[wave][m * 16 + l16][0];
        #pragma unroll
        for (int k = 0; k < 3; ++k) {
            const int kk = k * 4 + khalf;
            afrag[m][k].x = fr[kk];
            afrag[m][k].y = fr[kk + 1];
        }
    }

    // Accumulators initialized to broadcast bias (N = n*16 + l16 for all 8 rows)
    v8f acc[2][4];
    #pragma unroll
    for (int n = 0; n < 4; ++n) {
        const float bv = bias[n * 16 + l16];
        #pragma unroll
        for (int m = 0; m < 2; ++m)
            #pragma unroll
            for (int r = 0; r < 8; ++r) acc[m][n][r] = bv;
    }

    // ---- Phase 5: 24x V_WMMA_F32_16X16X4_F32 ------------------------------
    #pragma unroll
    for (int m = 0; m < 2; ++m)
        #pragma unroll
        for (int n = 0; n < 4; ++n)
            #pragma unroll
            for (int k = 0; k < 3; ++k)
                acc[m][n] = __builtin_amdgcn_wmma_f32_16x16x4_f32(
                    false, afrag[m][k], false, bfrag[n][k],
                    (short)0, acc[m][n], false, false);

    // ---- Phase 6: store D tiles: lane l -> M = r + 8*(l>=16), N = l%16 ----
    float* op = out + (size_t)b * 2048;     // 32*64 per board
    const int rowoff = (lane >> 4) * 8;
    #pragma unroll
    for (int m = 0; m < 2; ++m)
        #pragma unroll
        for (int r = 0; r < 8; ++r) {
            const int prow = m * 16 + rowoff + r;
            #pragma unroll
            for (int n = 0; n < 4; ++n)
                op[prow * 64 + n * 16 + l16] = acc[m][n][r];
        }
}

extern "C" void kernel_launch(void* const* d_in, const int* in_sizes, int n_in,
                              void* d_out, int out_size, void* d_ws, size_t ws_size,
                              hipStream_t stream) {
    const float* board = (const float*)d_in[0];   // [B,11,8,8]
    const int*   pids  = (const int*)  d_in[1];   // [B,8,8]
    const float* W     = (const float*)d_in[2];   // [64,11]
    const float* bias  = (const float*)d_in[3];   // [64]
    float* out = (float*)d_out;                   // [B,32,64]

    const int B = in_sizes[1] / 64;
    dim3 grid((B + WAVES_PER_BLOCK - 1) / WAVES_PER_BLOCK);
    piece_extract_gemm_kernel<<<grid, 256, 0, stream>>>(board, pids, W, bias, out, B);
}